// TGCN_18245021073500
// MI455X (gfx1250) — compile-verified
//
#include <hip/hip_runtime.h>
#include <cmath>

typedef __attribute__((ext_vector_type(2))) float v2f;
typedef __attribute__((ext_vector_type(8))) float v8f;

#define NNODES 50000
#define NEDGES 800000
#define F 128
#define NF (NNODES * F)   /* 6,400,000 floats per feature matrix */

// ---------------------------------------------------------------------------
// Degree / normalization helpers
// ---------------------------------------------------------------------------
__global__ void init_deg_kernel(float* __restrict__ deg) {
    int i = blockIdx.x * blockDim.x + threadIdx.x;
    if (i < NNODES) deg[i] = 1.0f;               // self-loop weight
}

__global__ void deg_accum_kernel(const int* __restrict__ tgt,
                                 const float* __restrict__ w,
                                 float* __restrict__ deg) {
    int i = blockIdx.x * blockDim.x + threadIdx.x;
    if (i < NEDGES) unsafeAtomicAdd(&deg[tgt[i]], w[i]);
}

__global__ void dinv_kernel(float* __restrict__ deg) {
    int i = blockIdx.x * blockDim.x + threadIdx.x;
    if (i < NNODES) deg[i] = rsqrtf(deg[i]);
}

// ---------------------------------------------------------------------------
// WMMA GEMM: out = epilogue(A1@W1 [+ A2@W2] + bias)
//   MODE 0: out = v                       (lin = X@W, no bias)
//   MODE 1: out = sigmoid(v)              (Z gate)
//   MODE 2: out = H * sigmoid(v)          (HR = H*R)
//   MODE 3: out = Z*H + (1-Z)*tanh(v)     (final combine)
// One wave (32 lanes) computes one 16x16 tile via V_WMMA_F32_16X16X4_F32.
// Block = 256 threads = 8 waves = the 8 column tiles of one 16-row strip.
// ---------------------------------------------------------------------------
template <int MODE, bool DUAL>
__global__ void wmma_gemm_kernel(const float* __restrict__ A1,
                                 const float* __restrict__ W1,
                                 const float* __restrict__ A2,
                                 const float* __restrict__ W2,
                                 const float* __restrict__ bias,
                                 const float* __restrict__ Hm,
                                 const float* __restrict__ Zm,
                                 float* __restrict__ out) {
    const int lane = threadIdx.x & 31;
    const int wave = threadIdx.x >> 5;
    const int half = lane >> 4;      // 0: lanes 0-15, 1: lanes 16-31
    const int l16  = lane & 15;
    const int m0   = blockIdx.x * 16;
    const int col  = wave * 16 + l16;
    const int row  = m0 + l16;

    v8f acc = {};

    // A 16x4 layout: lanes 0-15 hold K={k,k+1}, lanes 16-31 hold K={k+2,k+3}
    // B 4x16 layout: mirrored -- lane half selects K pair, lane%16 = N column
    {
        const float* ap = A1 + (size_t)row * F;
        #pragma unroll
        for (int k = 0; k < F; k += 4) {
            const int ka = k + 2 * half;
            v2f a; a.x = ap[ka]; a.y = ap[ka + 1];
            v2f b; b.x = W1[(size_t)ka * F + col];
                   b.y = W1[(size_t)(ka + 1) * F + col];
            acc = __builtin_amdgcn_wmma_f32_16x16x4_f32(
                false, a, false, b, (short)0, acc, false, false);
        }
    }
    if (DUAL) {
        const float* ap = A2 + (size_t)row * F;
        #pragma unroll
        for (int k = 0; k < F; k += 4) {
            const int ka = k + 2 * half;
            v2f a; a.x = ap[ka]; a.y = ap[ka + 1];
            v2f b; b.x = W2[(size_t)ka * F + col];
                   b.y = W2[(size_t)(ka + 1) * F + col];
            acc = __builtin_amdgcn_wmma_f32_16x16x4_f32(
                false, a, false, b, (short)0, acc, false, false);
        }
    }

    const float bv = (MODE == 0) ? 0.0f : bias[col];

    // C/D layout: VGPR j, lane half h -> M = j + 8*h, N = lane%16
    #pragma unroll
    for (int j = 0; j < 8; ++j) {
        const int r = m0 + j + 8 * half;
        const size_t o = (size_t)r * F + col;
        const float v = acc[j] + bv;
        if (MODE == 0) {
            out[o] = v;
        } else if (MODE == 1) {
            out[o] = 1.0f / (1.0f + __expf(-v));
        } else if (MODE == 2) {
            out[o] = Hm[o] * (1.0f / (1.0f + __expf(-v)));
        } else {
            const float z = Zm[o];
            out[o] = z * Hm[o] + (1.0f - z) * tanhf(v);
        }
    }
}

// ---------------------------------------------------------------------------
// agg_c = lin_c * dinv^2 + b_c   (self-loop term), all 3 convs at once
// ---------------------------------------------------------------------------
__global__ void init_agg_kernel(const float* __restrict__ lin,
                                const float* __restrict__ dinv,
                                const float* __restrict__ b_z,
                                const float* __restrict__ b_r,
                                const float* __restrict__ b_h,
                                float* __restrict__ agg) {
    const int idx = blockIdx.x * blockDim.x + threadIdx.x;   // over NF
    const int i = idx >> 7;
    const int f = idx & (F - 1);
    const float d = dinv[i];
    const float d2 = d * d;
    agg[idx]               = lin[idx]               * d2 + b_z[f];
    agg[(size_t)NF + idx]  = lin[(size_t)NF + idx]  * d2 + b_r[f];
    agg[(size_t)2*NF + idx]= lin[(size_t)2*NF + idx]* d2 + b_h[f];
}

// ---------------------------------------------------------------------------
// Edge scatter: 32 lanes per edge, 4 features/lane, all 3 convs fused.
// agg_c[tgt] += lin_c[src] * (dinv[src]*w*dinv[tgt])
// ---------------------------------------------------------------------------
__global__ void edge_scatter_kernel(const int* __restrict__ src,
                                    const int* __restrict__ tgt,
                                    const float* __restrict__ w,
                                    const float* __restrict__ dinv,
                                    const float* __restrict__ lin,
                                    float* __restrict__ agg) {
    const int idx  = blockIdx.x * blockDim.x + threadIdx.x;
    const int e    = idx >> 5;
    const int lane = idx & 31;
    if (e >= NEDGES) return;
    const int s = src[e];
    const int t = tgt[e];
    const float nw = dinv[s] * w[e] * dinv[t];
    const int f = lane << 2;
    const size_t so = (size_t)s * F + f;
    const size_t to = (size_t)t * F + f;
    #pragma unroll
    for (int c = 0; c < 3; ++c) {
        const size_t off = (size_t)c * NF;
        const float4 v = *(const float4*)(lin + off + so);
        float* ap = agg + off + to;
        unsafeAtomicAdd(ap + 0, v.x * nw);
        unsafeAtomicAdd(ap + 1, v.y * nw);
        unsafeAtomicAdd(ap + 2, v.z * nw);
        unsafeAtomicAdd(ap + 3, v.w * nw);
    }
}

// ---------------------------------------------------------------------------
extern "C" void kernel_launch(void* const* d_in, const int* in_sizes, int n_in,
                              void* d_out, int out_size, void* d_ws, size_t ws_size,
                              hipStream_t stream) {
    const float* X    = (const float*)d_in[0];
    const int*   eidx = (const int*)  d_in[1];   // [2, E]
    const float* ew   = (const float*)d_in[2];
    const float* H    = (const float*)d_in[3];
    const float* W_z  = (const float*)d_in[4];
    const float* b_z  = (const float*)d_in[5];
    const float* W_r  = (const float*)d_in[6];
    const float* b_r  = (const float*)d_in[7];
    const float* W_h  = (const float*)d_in[8];
    const float* b_h  = (const float*)d_in[9];
    const float* Wl_z = (const float*)d_in[10];
    const float* bl_z = (const float*)d_in[11];
    const float* Wl_r = (const float*)d_in[12];
    const float* bl_r = (const float*)d_in[13];
    const float* Wl_h = (const float*)d_in[14];
    const float* bl_h = (const float*)d_in[15];

    const int* src = eidx;
    const int* tgt = eidx + NEDGES;

    // Workspace layout (floats): dinv | lin[3*NF] | agg[3*NF]
    float* ws   = (float*)d_ws;
    float* dinv = ws;                         // NNODES (50000, %4==0)
    float* lin  = ws + NNODES;                // 3*NF
    float* agg  = lin + (size_t)3 * NF;       // 3*NF
    float* Zbuf = lin;                        // reuse (lin dead after scatter)
    float* HR   = lin + (size_t)NF;           // reuse
    float* out  = (float*)d_out;

    const int BT = 256;
    const int gemm_blocks = NNODES / 16;      // 3125, 8 waves cover all 128 cols

    // 1. degrees -> dinv
    init_deg_kernel<<<(NNODES + BT - 1) / BT, BT, 0, stream>>>(dinv);
    deg_accum_kernel<<<NEDGES / BT, BT, 0, stream>>>(tgt, ew, dinv);
    dinv_kernel<<<(NNODES + BT - 1) / BT, BT, 0, stream>>>(dinv);

    // 2. lin_c = X @ W_c   (fp32 WMMA)
    wmma_gemm_kernel<0, false><<<gemm_blocks, BT, 0, stream>>>(
        X, W_z, nullptr, nullptr, nullptr, nullptr, nullptr, lin);
    wmma_gemm_kernel<0, false><<<gemm_blocks, BT, 0, stream>>>(
        X, W_r, nullptr, nullptr, nullptr, nullptr, nullptr, lin + (size_t)NF);
    wmma_gemm_kernel<0, false><<<gemm_blocks, BT, 0, stream>>>(
        X, W_h, nullptr, nullptr, nullptr, nullptr, nullptr, lin + (size_t)2 * NF);

    // 3. agg_c = lin_c*dinv^2 + b_c  (self-loop) then edge scatter-add
    init_agg_kernel<<<NF / BT, BT, 0, stream>>>(lin, dinv, b_z, b_r, b_h, agg);
    edge_scatter_kernel<<<(NEDGES * 32) / BT, BT, 0, stream>>>(
        src, tgt, ew, dinv, lin, agg);

    // 4. Z = sigmoid(cz@Wlz_top + H@Wlz_bot + bl_z)
    wmma_gemm_kernel<1, true><<<gemm_blocks, BT, 0, stream>>>(
        agg, Wl_z, H, Wl_z + F * F, bl_z, nullptr, nullptr, Zbuf);

    // 5. HR = H * sigmoid(cr@Wlr_top + H@Wlr_bot + bl_r)
    wmma_gemm_kernel<2, true><<<gemm_blocks, BT, 0, stream>>>(
        agg + (size_t)NF, Wl_r, H, Wl_r + F * F, bl_r, H, nullptr, HR);

    // 6. out = Z*H + (1-Z)*tanh(ch@Wlh_top + HR@Wlh_bot + bl_h)
    wmma_gemm_kernel<3, true><<<gemm_blocks, BT, 0, stream>>>(
        agg + (size_t)2 * NF, Wl_h, HR, Wl_h + F * F, bl_h, H, Zbuf, out);
}